// InteractionModuleEmbed_11347303596603
// MI455X (gfx1250) — compile-verified
//
#include <hip/hip_runtime.h>
#include <math.h>

typedef __attribute__((ext_vector_type(2))) float v2f;
typedef __attribute__((ext_vector_type(8))) float v8f;

constexpr int B = 16, N_VEC = 2048, N_DIM = 32, N_W = 4, N_HID = 128, N_CLASS = 10;
constexpr int HSTRIDE = N_HID + 4;  // LDS padding to avoid bank conflicts

__device__ __forceinline__ float gelu_exact(float x) {
  return 0.5f * x * (1.0f + erff(x * 0.7071067811865475f));
}

// X[b,n,:] = emb[data[b,n],:]
__global__ __launch_bounds__(256) void gather_kernel(const int* __restrict__ data,
                                                     const float* __restrict__ emb,
                                                     float* __restrict__ X) {
  int idx = blockIdx.x * blockDim.x + threadIdx.x;
  if (idx >= B * N_VEC * N_DIM) return;
  int row = idx / N_DIM;
  int d = idx - row * N_DIM;
  X[idx] = emb[data[row] * N_DIM + d];
}

// T[b] = fw2_i (128 x 2048) @ V[b] (2048 x 32).  One wave per (b, mtile, ntile).
__global__ __launch_bounds__(32) void t_kernel(const float* __restrict__ fw2_i,
                                               const float* __restrict__ Vin,
                                               float* __restrict__ T) {
  const int b = blockIdx.x, mt = blockIdx.y, nt = blockIdx.z;
  const int lane = threadIdx.x;
  const int half = lane >> 4;   // K sub-select (0 -> K{0,1}, 1 -> K{2,3})
  const int l = lane & 15;      // row (A) / col (B) index within tile
  const float* Vb = Vin + b * N_VEC * N_DIM;
  const int mrow = mt * 16 + l;
  const int ncol = nt * 16 + l;
  v8f c = {};
  for (int k = 0; k < N_VEC; k += 4) {
    const int ka = k + 2 * half;
    v2f a = *(const v2f*)(fw2_i + mrow * N_VEC + ka);  // contiguous 8B load
    v2f bb;
    bb.x = Vb[ka * N_DIM + ncol];
    bb.y = Vb[(ka + 1) * N_DIM + ncol];
    c = __builtin_amdgcn_wmma_f32_16x16x4_f32(false, a, false, bb, (short)0, c,
                                              false, false);
  }
  float* Tb = T + b * N_HID * N_DIM;
#pragma unroll
  for (int r = 0; r < 8; ++r)
    Tb[(mt * 16 + r + 8 * half) * N_DIM + nt * 16 + l] = c[r];
}

// s[b,d] = sum_m fb2_i[m] * V[b,m,d]
__global__ __launch_bounds__(256) void s_kernel(const float* __restrict__ fb2_i,
                                                const float* __restrict__ Vin,
                                                float* __restrict__ s) {
  int idx = blockIdx.x * blockDim.x + threadIdx.x;
  if (idx >= B * N_DIM) return;
  int b = idx / N_DIM, d = idx - b * N_DIM;
  const float* Vb = Vin + b * N_VEC * N_DIM;
  float acc = 0.f;
  for (int m = 0; m < N_VEC; ++m) acc += fb2_i[m] * Vb[m * N_DIM + d];
  s[idx] = acc;
}

// One wave per 16-row tile:
//   h = gelu(X_tile @ fw1_i + fb1_i)   (16x128, staged in LDS)
//   Vout_tile = h @ T[b] + s[b]        (16x32)
__global__ __launch_bounds__(32) void fused_layer_kernel(
    const float* __restrict__ X, const float* __restrict__ fw1_i,
    const float* __restrict__ fb1_i, const float* __restrict__ T,
    const float* __restrict__ s, float* __restrict__ Vout) {
  __shared__ float hs[16 * HSTRIDE];
  const int tile = blockIdx.x;
  const int b = tile / (N_VEC / 16);
  const int n0 = (tile % (N_VEC / 16)) * 16;
  const int lane = threadIdx.x;
  const int half = lane >> 4;
  const int l = lane & 15;

  // ---- phase 1: h = gelu(X @ fw1 + fb1), K = 32 ----
  const float* Xrow = X + (size_t)(b * N_VEC + n0 + l) * N_DIM;
  for (int nt = 0; nt < N_HID / 16; ++nt) {
    const int ncol = nt * 16 + l;
    v8f c = {};
#pragma unroll
    for (int k = 0; k < N_DIM; k += 4) {
      const int ka = k + 2 * half;
      v2f a = *(const v2f*)(Xrow + ka);
      v2f bb;
      bb.x = fw1_i[ka * N_HID + ncol];
      bb.y = fw1_i[(ka + 1) * N_HID + ncol];
      c = __builtin_amdgcn_wmma_f32_16x16x4_f32(false, a, false, bb, (short)0, c,
                                                false, false);
    }
    const float bias = fb1_i[ncol];
#pragma unroll
    for (int r = 0; r < 8; ++r)
      hs[(r + 8 * half) * HSTRIDE + ncol] = gelu_exact(c[r] + bias);
  }
  __syncthreads();

  // ---- phase 2: Vout = h @ T + s, K = 128 ----
  const float* Tb = T + b * N_HID * N_DIM;
  const float* sb = s + b * N_DIM;
  for (int nt = 0; nt < N_DIM / 16; ++nt) {
    const int ncol = nt * 16 + l;
    v8f c = {};
    for (int k = 0; k < N_HID; k += 4) {
      const int ka = k + 2 * half;
      v2f a = *(const v2f*)(&hs[l * HSTRIDE + ka]);  // ds_load_b64
      v2f bb;
      bb.x = Tb[ka * N_DIM + ncol];
      bb.y = Tb[(ka + 1) * N_DIM + ncol];
      c = __builtin_amdgcn_wmma_f32_16x16x4_f32(false, a, false, bb, (short)0, c,
                                                false, false);
    }
    const float sv = sb[ncol];
#pragma unroll
    for (int r = 0; r < 8; ++r)
      Vout[(size_t)(b * N_VEC + n0 + r + 8 * half) * N_DIM + ncol] = c[r] + sv;
  }
}

// out[b,c] = V[b,:].flatten() @ Wf[:,c] + bf[c]
__global__ __launch_bounds__(256) void final_kernel(const float* __restrict__ V,
                                                    const float* __restrict__ Wf,
                                                    const float* __restrict__ bf,
                                                    float* __restrict__ out) {
  __shared__ float red[N_CLASS][256];
  const int b = blockIdx.x;
  const int tid = threadIdx.x;
  float acc[N_CLASS];
#pragma unroll
  for (int c = 0; c < N_CLASS; ++c) acc[c] = 0.f;
  const float* Vb = V + (size_t)b * N_VEC * N_DIM;
  for (int k = tid; k < N_VEC * N_DIM; k += 256) {
    const float v = Vb[k];
    const float* w = Wf + (size_t)k * N_CLASS;
#pragma unroll
    for (int c = 0; c < N_CLASS; ++c) acc[c] += v * w[c];
  }
#pragma unroll
  for (int c = 0; c < N_CLASS; ++c) red[c][tid] = acc[c];
  __syncthreads();
  for (int off = 128; off > 0; off >>= 1) {
    if (tid < off) {
#pragma unroll
      for (int c = 0; c < N_CLASS; ++c) red[c][tid] += red[c][tid + off];
    }
    __syncthreads();
  }
  if (tid == 0) {
#pragma unroll
    for (int c = 0; c < N_CLASS; ++c) out[b * N_CLASS + c] = red[c][0] + bf[c];
  }
}

extern "C" void kernel_launch(void* const* d_in, const int* in_sizes, int n_in,
                              void* d_out, int out_size, void* d_ws, size_t ws_size,
                              hipStream_t stream) {
  (void)in_sizes; (void)n_in; (void)out_size; (void)ws_size;
  const int*   data = (const int*)d_in[0];
  const float* emb  = (const float*)d_in[1];
  const float* fw1  = (const float*)d_in[2];
  const float* fb1  = (const float*)d_in[3];
  const float* fw2  = (const float*)d_in[4];
  const float* fb2  = (const float*)d_in[5];
  const float* Wf   = (const float*)d_in[6];
  const float* bf   = (const float*)d_in[7];

  constexpr size_t XSZ = (size_t)B * N_VEC * N_DIM;
  float* X  = (float*)d_ws;
  float* V0 = X + XSZ;
  float* V1 = V0 + XSZ;
  float* T  = V1 + XSZ;                       // B * 128 * 32
  float* s  = T + (size_t)B * N_HID * N_DIM;  // B * 32

  gather_kernel<<<(B * N_VEC * N_DIM + 255) / 256, 256, 0, stream>>>(data, emb, X);

  const float* Vin = X;   // first layer consumes X directly (V starts equal to X)
  float* bufs[2] = {V0, V1};
  int cur = 0;
  for (int i = N_W - 1; i >= 0; --i) {
    float* Vout = bufs[cur];
    dim3 tg(B, N_HID / 16, N_DIM / 16);
    t_kernel<<<tg, 32, 0, stream>>>(fw2 + (size_t)i * N_HID * N_VEC, Vin, T);
    s_kernel<<<(B * N_DIM + 255) / 256, 256, 0, stream>>>(fb2 + (size_t)i * N_VEC, Vin, s);
    fused_layer_kernel<<<B * N_VEC / 16, 32, 0, stream>>>(
        X, fw1 + (size_t)i * N_DIM * N_HID, fb1 + (size_t)i * N_HID, T, s, Vout);
    Vin = Vout;
    cur ^= 1;
  }

  final_kernel<<<B, 256, 0, stream>>>(Vin, Wf, bf, (float*)d_out);
}